// UncertaintyLoss_47244640256380
// MI455X (gfx1250) — compile-verified
//
#include <hip/hip_runtime.h>
#include <hip/hip_bf16.h>

typedef __attribute__((ext_vector_type(2))) float v2f;
typedef __attribute__((ext_vector_type(8))) float v8f;

#define NHEAD 3
#define B 384
#define D 128
#define S 25
#define MARGIN 1.0f

// ---------------- workspace layout (floats) ----------------
// e_norm : NHEAD*B*D       = 147456
// dist   : NHEAD*B*B       = 442368
// sq     : NHEAD*B         = 1152
// lab2   : B               = 384
// psum   : NHEAD*B         = 1152
// pcnt   : NHEAD*B         = 1152
#define OFF_ENORM 0
#define OFF_DIST  (OFF_ENORM + NHEAD*B*D)
#define OFF_SQ    (OFF_DIST  + NHEAD*B*B)
#define OFF_LAB2  (OFF_SQ    + NHEAD*B)
#define OFF_PSUM  (OFF_LAB2  + B)
#define OFF_PCNT  (OFF_PSUM  + NHEAD*B)

// Kernel 1: lab2[b] = mean over s of target[b, s, 2];  target is (B, S, 3)
__global__ void k_lab2(const float* __restrict__ target, float* __restrict__ lab2) {
    int b = threadIdx.x;  // blockDim = 384
    if (b < B) {
        float s = 0.0f;
        #pragma unroll
        for (int j = 0; j < S; ++j) s += target[b * (S * NHEAD) + j * NHEAD + 2];
        lab2[b] = s * (1.0f / (float)S);
    }
}

// Kernel 2: row-normalize embeddings, emit sq[row] = sum(e_norm^2).
// One wave (32 lanes) per row; 4 floats per lane.
__global__ void k_normalize(const float* __restrict__ emb, float* __restrict__ e_norm,
                            float* __restrict__ sqv) {
    int wave = (blockIdx.x * blockDim.x + threadIdx.x) >> 5;   // global wave id
    int lane = threadIdx.x & 31;
    if (wave >= NHEAD * B) return;
    const float* row = emb + (size_t)wave * D;
    float v0 = row[lane +  0], v1 = row[lane + 32], v2 = row[lane + 64], v3 = row[lane + 96];
    float ss = v0*v0 + v1*v1 + v2*v2 + v3*v3;
    #pragma unroll
    for (int off = 16; off > 0; off >>= 1) ss += __shfl_xor(ss, off, 32);
    float norm = sqrtf(ss);
    float inv  = 1.0f / fmaxf(norm, 1e-12f);
    float* orow = e_norm + (size_t)wave * D;
    orow[lane +  0] = v0 * inv;
    orow[lane + 32] = v1 * inv;
    orow[lane + 64] = v2 * inv;
    orow[lane + 96] = v3 * inv;
    if (lane == 0) sqv[wave] = ss * inv * inv;   // sum of squares of normalized row
}

// Kernel 3: Gram via V_WMMA_F32_16X16X4_F32, fused distance epilogue.
// One wave per 16x16 output tile. 24x24 tiles per head, 3 heads = 1728 tiles.
// A-tile layout (32-bit 16x4): lanes 0-15 -> M rows, VGPR0/1 = K,K+1; lanes 16-31 -> K+2,K+3.
// B-tile (4x16, Bᵀ of same matrix) loads identically with the column-tile row base.
__global__ void k_gram_dist(const float* __restrict__ e_norm, const float* __restrict__ sqv,
                            float* __restrict__ dist) {
    int wave = (blockIdx.x * blockDim.x + threadIdx.x) >> 5;
    int lane = threadIdx.x & 31;
    const int tiles_per_head = (B / 16) * (B / 16);  // 576
    if (wave >= NHEAD * tiles_per_head) return;

    int head = wave / tiles_per_head;
    int rem  = wave % tiles_per_head;
    int m0 = (rem / (B / 16)) * 16;
    int n0 = (rem % (B / 16)) * 16;

    int half = lane >> 4;        // 0: K,K+1   1: K+2,K+3
    int l16  = lane & 15;

    const float* eh   = e_norm + (size_t)head * B * D;
    const float* arow = eh + (size_t)(m0 + l16) * D;
    const float* brow = eh + (size_t)(n0 + l16) * D;

    v8f acc = {};
    #pragma unroll 8
    for (int k = 0; k < D; k += 4) {
        int kk = k + 2 * half;
        v2f a = *(const v2f*)(arow + kk);
        v2f b = *(const v2f*)(brow + kk);
        // 8 args: (neg_a, A, neg_b, B, c_mod, C, reuse_a, reuse_b)
        acc = __builtin_amdgcn_wmma_f32_16x16x4_f32(
            false, a, false, b, (short)0, acc, false, false);
    }

    // Epilogue: C/D layout — VGPR r: lanes 0-15 -> M=r, lanes 16-31 -> M=8+r; N = lane%16.
    const float* sqh = sqv + head * B;
    float sqb = sqh[n0 + l16];
    float* dh = dist + (size_t)head * B * B;
    #pragma unroll
    for (int r = 0; r < 8; ++r) {
        int mrow = m0 + r + 8 * half;
        float g  = acc[r];
        float d2 = sqh[mrow] + sqb - 2.0f * g;
        dh[(size_t)mrow * B + (n0 + l16)] = sqrtf(fmaxf(d2, 0.0f));
    }
}

// Kernel 4: semihard triplet mining. One block per (head, anchor).
// Each block deterministically writes its own partial (sum, count) slot.
__global__ void k_triplet(const float* __restrict__ dist, const int* __restrict__ label_item,
                          const float* __restrict__ lab2, float* __restrict__ psum,
                          float* __restrict__ pcnt) {
    __shared__ float drow[B];
    __shared__ float lab[B];
    __shared__ float rs[256];
    __shared__ float rc[256];

    int bid  = blockIdx.x;           // 0 .. NHEAD*B-1
    int head = bid / B;
    int a    = bid % B;
    int tid  = threadIdx.x;          // blockDim = 256

    const float* dh = dist + (size_t)head * B * B + (size_t)a * B;
    for (int t = tid; t < B; t += 256) {
        drow[t] = dh[t];
        lab[t]  = (head == 2) ? lab2[t] : (float)label_item[t];
    }
    __syncthreads();

    float la  = lab[a];
    float sum = 0.0f;
    float cnt = 0.0f;
    for (int idx = tid; idx < B * B; idx += 256) {
        int p = idx / B;
        int n = idx % B;
        if (p != a && lab[p] == la && lab[n] != la) {
            float tm = drow[n] - drow[p];           // an - ap
            if (tm > 0.0f && tm < MARGIN) {         // mined & loss_v > 0
                sum += MARGIN - tm;
                cnt += 1.0f;
            }
        }
    }
    rs[tid] = sum; rc[tid] = cnt;
    __syncthreads();
    for (int s2 = 128; s2 > 0; s2 >>= 1) {
        if (tid < s2) { rs[tid] += rs[tid + s2]; rc[tid] += rc[tid + s2]; }
        __syncthreads();
    }
    if (tid == 0) { psum[bid] = rs[0]; pcnt[bid] = rc[0]; }
}

// Kernel 5: reduce partials per head, write [loss_sum, l0, l1, l2].
__global__ void k_finalize(const float* __restrict__ psum, const float* __restrict__ pcnt,
                           float* __restrict__ out) {
    __shared__ float rs[128];
    __shared__ float rc[128];
    __shared__ float res[NHEAD];
    int tid = threadIdx.x;           // blockDim = 128
    for (int h = 0; h < NHEAD; ++h) {
        float s = 0.0f, c = 0.0f;
        for (int t = tid; t < B; t += 128) { s += psum[h * B + t]; c += pcnt[h * B + t]; }
        rs[tid] = s; rc[tid] = c;
        __syncthreads();
        for (int s2 = 64; s2 > 0; s2 >>= 1) {
            if (tid < s2) { rs[tid] += rs[tid + s2]; rc[tid] += rc[tid + s2]; }
            __syncthreads();
        }
        if (tid == 0) res[h] = (rc[0] > 0.0f) ? rs[0] / fmaxf(rc[0], 1.0f) : 0.0f;
        __syncthreads();
    }
    if (tid == 0) {
        out[0] = res[0] + res[1] + res[2];
        out[1] = res[0];
        out[2] = res[1];
        out[3] = res[2];
    }
}

extern "C" void kernel_launch(void* const* d_in, const int* in_sizes, int n_in,
                              void* d_out, int out_size, void* d_ws, size_t ws_size,
                              hipStream_t stream) {
    const float* vr_classes = (const float*)d_in[1];   // (3, 384, 128) f32
    const float* target     = (const float*)d_in[2];   // (384, 25, 3)  f32
    const int*   label_item = (const int*)d_in[4];     // (384,)        i32
    float* out = (float*)d_out;

    float* ws     = (float*)d_ws;
    float* e_norm = ws + OFF_ENORM;
    float* dist   = ws + OFF_DIST;
    float* sqv    = ws + OFF_SQ;
    float* lab2   = ws + OFF_LAB2;
    float* psum   = ws + OFF_PSUM;
    float* pcnt   = ws + OFF_PCNT;

    // 1) head-2 labels
    k_lab2<<<1, B, 0, stream>>>(target, lab2);

    // 2) normalize rows: 1152 waves, 4 waves/block (128 thr) -> 288 blocks
    k_normalize<<<(NHEAD * B) / 4, 128, 0, stream>>>(vr_classes, e_norm, sqv);

    // 3) WMMA gram + distance: 1728 tile-waves, 8 waves/block (256 thr) -> 216 blocks
    k_gram_dist<<<(NHEAD * (B/16) * (B/16)) / 8, 256, 0, stream>>>(e_norm, sqv, dist);

    // 4) triplet mining: one block per (head, anchor)
    k_triplet<<<NHEAD * B, 256, 0, stream>>>(dist, label_item, lab2, psum, pcnt);

    // 5) finalize
    k_finalize<<<1, 128, 0, stream>>>(psum, pcnt, out);
}